// TransportOperatorHeader_64939905515616
// MI455X (gfx1250) — compile-verified
//
#include <hip/hip_runtime.h>
#include <cmath>

// ---------------- problem constants (mirror reference) ----------------
#define B_      256     // batch
#define D_      512     // feature dim
#define N_      64      // block dim (operator size)
#define KB_     8       // blocks per feature vector (512/64)
#define M_      64      // dictionary size
#define LAM_    0.04f
#define LR_     0.01f
#define FISTA_ITERS 15
#define TAYLOR_N    6   // Taylor-Horner order for expm
#define NSQUARE     3   // scaling-and-squaring steps
// Padded LDS row stride: 68 mod 64 = 4 -> A-fragment (row-varying) reads hit
// distinct banks, B-column reads are lane-consecutive, rows stay 16B-aligned
// (b128/b64 LDS ops legal), and A fragment pairs are 8B-aligned (b64 loads).
#define LDSTR   68

typedef float v2f __attribute__((ext_vector_type(2)));
typedef float v8f __attribute__((ext_vector_type(8)));

// ---------------------------------------------------------------------
// 64x64 f32 matmul in LDS on V_WMMA_F32_16X16X4_F32.
//   C = alpha * (A1*B1 [+ A2*B2])  (+ I on the diagonal if addI)
// 8 waves, 16 tiles of 16x16, 2 tiles/wave sharing tj (so B fragments are
// loaded once per wave).  All fragments are prefetched into register arrays
// before the dependent WMMA chain so the DS loads issue as a burst and the
// dscnt waits pipeline instead of stalling at 0 per pair.
// ---------------------------------------------------------------------
template <bool FUSED>
__device__ __forceinline__ void mm64t(float* __restrict__ C,
                                      const float* __restrict__ A1,
                                      const float* __restrict__ B1,
                                      const float* __restrict__ A2,
                                      const float* __restrict__ B2,
                                      float alpha, bool addI, int tid) {
  const int lane  = tid & 31;
  const int wave  = tid >> 5;
  const int lrow  = lane & 15;
  const int khalf = (lane >> 4) << 1;   // 0 or 2
  const int tj    = wave & 3;           // both tiles of this wave share tj
  const int ti0   = wave >> 2;          // tile rows ti0 and ti0+2
  const int n     = tj * 16 + lrow;

  // ---- B fragments (shared across both tiles) ----
  v2f b1[16];
  {
    const float* Bcol = B1 + n;
#pragma unroll
    for (int kk = 0; kk < 16; ++kk) {
      const int k0 = kk * 4 + khalf;
      b1[kk].x = Bcol[(k0)     * LDSTR];
      b1[kk].y = Bcol[(k0 + 1) * LDSTR];
    }
  }
  v2f b2[16];
  if (FUSED) {
    const float* Bcol = B2 + n;
#pragma unroll
    for (int kk = 0; kk < 16; ++kk) {
      const int k0 = kk * 4 + khalf;
      b2[kk].x = Bcol[(k0)     * LDSTR];
      b2[kk].y = Bcol[(k0 + 1) * LDSTR];
    }
  }

#pragma unroll
  for (int half = 0; half < 2; ++half) {
    const int ti = ti0 + 2 * half;
    v8f acc = {0.f, 0.f, 0.f, 0.f, 0.f, 0.f, 0.f, 0.f};
    {
      const float* Arow = A1 + (ti * 16 + lrow) * LDSTR + khalf;
      v2f a1[16];
#pragma unroll
      for (int kk = 0; kk < 16; ++kk)      // burst of b64 LDS loads
        a1[kk] = *(const v2f*)(Arow + kk * 4);
#pragma unroll
      for (int kk = 0; kk < 16; ++kk)
        acc = __builtin_amdgcn_wmma_f32_16x16x4_f32(
            false, a1[kk], false, b1[kk], (short)0, acc, false, false);
    }
    if (FUSED) {
      const float* Arow = A2 + (ti * 16 + lrow) * LDSTR + khalf;
      v2f a2[16];
#pragma unroll
      for (int kk = 0; kk < 16; ++kk)
        a2[kk] = *(const v2f*)(Arow + kk * 4);
#pragma unroll
      for (int kk = 0; kk < 16; ++kk)
        acc = __builtin_amdgcn_wmma_f32_16x16x4_f32(
            false, a2[kk], false, b2[kk], (short)0, acc, false, false);
    }
    const int rbase = ti * 16 + ((lane >> 4) << 3);
#pragma unroll
    for (int r = 0; r < 8; ++r) {
      const float d = (addI && (rbase + r) == n) ? 1.0f : 0.0f;
      C[(rbase + r) * LDSTR + n] = fmaf(alpha, acc[r], d);
    }
  }
}

// T = sum_m coeff[m] * psi[m].  Each thread owns 16 consecutive elements of
// one row (p = tid/4, q0 = (tid%4)*16): 4 b128 global loads per m, 4 b128
// LDS stores at the end.
__device__ __forceinline__ void assembleT(float* __restrict__ Tm,
                                          const float* __restrict__ psi,
                                          const float* __restrict__ coeff,
                                          int tid) {
  const int base = tid * 16;
  const int p  = tid >> 2;
  const int q0 = (tid & 3) * 16;
  float a[16];
#pragma unroll
  for (int j = 0; j < 16; ++j) a[j] = 0.f;
  const float* pp = psi + base;
  for (int m = 0; m < M_; ++m) {
    const float4* pv = (const float4*)(pp + (size_t)m * (N_ * N_));
    const float   cm = coeff[m];
#pragma unroll
    for (int j4 = 0; j4 < 4; ++j4) {
      const float4 x = pv[j4];
      a[4 * j4 + 0] += cm * x.x;
      a[4 * j4 + 1] += cm * x.y;
      a[4 * j4 + 2] += cm * x.z;
      a[4 * j4 + 3] += cm * x.w;
    }
  }
  float4* tv = (float4*)&Tm[p * LDSTR + q0];
#pragma unroll
  for (int j4 = 0; j4 < 4; ++j4) {
    float4 x;
    x.x = a[4 * j4 + 0]; x.y = a[4 * j4 + 1];
    x.z = a[4 * j4 + 2]; x.w = a[4 * j4 + 3];
    tv[j4] = x;
  }
}

// E = expm(Tm): order-6 Taylor-Horner on Tm/2^s, then s squarings.
// Loops fully unrolled so every alpha is a literal constant (no fp division
// in the hot path); +I folded into the WMMA write-back.
__device__ float* expm64(const float* __restrict__ Tm, float* Y, float* P,
                         int tid) {
  constexpr float inv2s = 1.0f / (float)(1 << NSQUARE);
#pragma unroll
  for (int r = 0; r < 16; ++r) {
    const int idx = tid + 256 * r;
    const int p = idx >> 6, q = idx & 63;
    Y[p * LDSTR + q] = (p == q) ? 1.f : 0.f;
  }
  __syncthreads();
#pragma unroll
  for (int k = TAYLOR_N; k >= 1; --k) {
    mm64t<false>(P, Tm, Y, nullptr, nullptr, inv2s / (float)k, true, tid);
    __syncthreads();
    float* tp = Y; Y = P; P = tp;
  }
#pragma unroll
  for (int s = 0; s < NSQUARE; ++s) {
    mm64t<false>(P, Y, Y, nullptr, nullptr, 1.f, false, tid);
    __syncthreads();
    float* tp = Y; Y = P; P = tp;
  }
  return Y;
}

// ---------------------------------------------------------------------
// One workgroup (256 threads = 8 wave32) per batch element.
// ---------------------------------------------------------------------
__global__ __launch_bounds__(256)
void transop_kernel(const float* __restrict__ z0g,
                    const float* __restrict__ z1g,
                    const float* __restrict__ psi,
                    float* __restrict__ out) {
  __shared__ float sMat[6][N_ * LDSTR];  // six 64x64 working matrices
  __shared__ float sZ0[KB_ * N_];
  __shared__ float sZ1[KB_ * N_];
  __shared__ float sR [KB_ * N_];
  __shared__ float sy[M_], sc[M_], sdc[M_];
  __shared__ float sRed[256];

  const int tid = threadIdx.x;
  const int b   = blockIdx.x;
  constexpr float inv2s = 1.0f / (float)(1 << NSQUARE);

  sZ0[tid]       = z0g[(size_t)b * D_ + tid];
  sZ0[tid + 256] = z0g[(size_t)b * D_ + 256 + tid];
  sZ1[tid]       = z1g[(size_t)b * D_ + tid];
  sZ1[tid + 256] = z1g[(size_t)b * D_ + 256 + tid];
  if (tid < M_) { sy[tid] = 0.f; sc[tid] = 0.f; }
  __syncthreads();

  float tmom = 1.0f;  // FISTA momentum (same deterministic scalar in all lanes)

  for (int it = 0; it < FISTA_ITERS; ++it) {
    // ---- T(y) ----
    float* T = sMat[0];
    assembleT(T, psi, sy, tid);
    __syncthreads();

    // ---- phase 1: E = expm(T) ----
    float* E = expm64(T, sMat[1], sMat[2], tid);

    // ---- residual R = E z0 - z1  (R[k,p] = sum_q E[p,q] z0[k,q] - z1[k,p])
    for (int r = 0; r < 2; ++r) {
      const int idx = tid + 256 * r;
      const int kb = idx >> 6, p = idx & 63;
      const float4* ev = (const float4*)&E[p * LDSTR];
      const float4* zv = (const float4*)&sZ0[kb * N_];
      float acc = 0.f;
#pragma unroll
      for (int q4 = 0; q4 < 16; ++q4) {
        const float4 e = ev[q4];
        const float4 z = zv[q4];
        acc += e.x * z.x + e.y * z.y + e.z * z.z + e.w * z.w;
      }
      sR[idx] = acc - sZ1[idx];
    }
    __syncthreads();

    // ---- Gbar = R^T Z0 (dL/dE) pre-scaled by 1/2^s; S = T^T / 2^s ----
    float* G = sMat[3];
    float* S = sMat[4];
    for (int r = 0; r < 16; ++r) {
      const int idx = tid + 256 * r;
      const int p = idx >> 6, q = idx & 63;
      float acc = 0.f;
#pragma unroll
      for (int kb = 0; kb < KB_; ++kb) acc += sR[kb * N_ + p] * sZ0[kb * N_ + q];
      G[p * LDSTR + q] = inv2s * acc;
      S[p * LDSTR + q] = inv2s * T[q * LDSTR + p];
    }
    __syncthreads();

    // ---- phase 2: dL/dT = L(T^T, Gbar) via block expm [[S,G],[0,S]] ----
    // Block-triangular with equal diagonals: each block product is one
    // single matmul (Pa) + one fused dual-product matmul (Pb).
    float* Ya = sMat[0];  // T dead (S built)
    float* Yb = sMat[1];  // expm scratch; E is sMat[2] (dead, R built)
    float* Pa = sMat[2];
    float* Pb = sMat[5];
#pragma unroll
    for (int r = 0; r < 16; ++r) {
      const int idx = tid + 256 * r;
      const int p = idx >> 6, q = idx & 63;
      Ya[p * LDSTR + q] = (p == q) ? 1.f : 0.f;
      Yb[p * LDSTR + q] = 0.f;
    }
    __syncthreads();
#pragma unroll
    for (int k = TAYLOR_N; k >= 1; --k) {
      const float a = 1.0f / (float)k;
      mm64t<false>(Pa, S, Ya, nullptr, nullptr, a, true, tid);  // I + S*Ya/k
      mm64t<true >(Pb, S, Yb, G, Ya, a, false, tid);            // (S*Yb+G*Ya)/k
      __syncthreads();
      float* t0 = Ya; Ya = Pa; Pa = t0;
      float* t1 = Yb; Yb = Pb; Pb = t1;
    }
#pragma unroll
    for (int s = 0; s < NSQUARE; ++s) {
      mm64t<false>(Pa, Ya, Ya, nullptr, nullptr, 1.f, false, tid); // E' = E*E
      mm64t<true >(Pb, Ya, Yb, Yb, Ya, 1.f, false, tid);           // L' = E*L+L*E
      __syncthreads();
      float* t0 = Ya; Ya = Pa; Pa = t0;
      float* t1 = Yb; Yb = Pb; Pb = t1;
    }
    // Yb holds dL/dT

    // ---- dc_m = <dL/dT, psi_m>  (b128 global + b128 LDS reads) ----
    {
      const int m = tid >> 2, part = tid & 3;
      const float4* pv = (const float4*)(psi + (size_t)m * (N_ * N_) + part * 1024);
      float acc = 0.f;
      for (int j4 = 0; j4 < 256; ++j4) {
        const int idx = part * 1024 + j4 * 4;
        const float4 x = pv[j4];
        const float4 y = *(const float4*)&Yb[(idx >> 6) * LDSTR + (idx & 63)];
        acc += x.x * y.x + x.y * y.y + x.z * y.z + x.w * y.w;
      }
      sRed[tid] = acc;
    }
    __syncthreads();
    if (tid < M_)
      sdc[tid] = sRed[4 * tid] + sRed[4 * tid + 1] +
                 sRed[4 * tid + 2] + sRed[4 * tid + 3];
    __syncthreads();

    // ---- FISTA update ----
    const float tnew = 0.5f * (1.0f + sqrtf(1.0f + 4.0f * tmom * tmom));
    if (tid < M_) {
      const float x   = sy[tid] - LR_ * sdc[tid];
      const float thr = LR_ * LAM_;
      const float cn  = (x > thr) ? (x - thr) : ((x < -thr) ? (x + thr) : 0.f);
      sy[tid] = cn + ((tmom - 1.0f) / tnew) * (cn - sc[tid]);
      sc[tid] = cn;
    }
    tmom = tnew;
    __syncthreads();
  }

  // ---- final: z1_hat = expm(T(c)) z0 ----
  float* T = sMat[0];
  assembleT(T, psi, sc, tid);
  __syncthreads();
  float* E = expm64(T, sMat[1], sMat[2], tid);
  for (int r = 0; r < 2; ++r) {
    const int idx = tid + 256 * r;
    const int kb = idx >> 6, p = idx & 63;
    const float4* ev = (const float4*)&E[p * LDSTR];
    const float4* zv = (const float4*)&sZ0[kb * N_];
    float acc = 0.f;
#pragma unroll
    for (int q4 = 0; q4 < 16; ++q4) {
      const float4 e = ev[q4];
      const float4 z = zv[q4];
      acc += e.x * z.x + e.y * z.y + e.z * z.z + e.w * z.w;
    }
    out[(size_t)b * D_ + idx] = acc;
  }
}

extern "C" void kernel_launch(void* const* d_in, const int* in_sizes, int n_in,
                              void* d_out, int out_size, void* d_ws, size_t ws_size,
                              hipStream_t stream) {
  const float* z0  = (const float*)d_in[0];
  const float* z1  = (const float*)d_in[1];
  const float* psi = (const float*)d_in[2];
  float* out = (float*)d_out;
  (void)in_sizes; (void)n_in; (void)out_size; (void)d_ws; (void)ws_size;
  hipLaunchKernelGGL(transop_kernel, dim3(B_), dim3(256), 0, stream,
                     z0, z1, psi, out);
}